// SVDLoraWrapper_91199335563893
// MI455X (gfx1250) — compile-verified
//
#include <hip/hip_runtime.h>
#include <hip/hip_bf16.h>

typedef float v2f __attribute__((ext_vector_type(2)));
typedef float v4f __attribute__((ext_vector_type(4)));
typedef float v8f __attribute__((ext_vector_type(8)));
typedef int   v4i __attribute__((ext_vector_type(4)));

#define BM 128
#define BN 128
#define BK 32
#define LDT 36   // BK + 4 pad: 144B row stride (16B aligned), bank-conflict-free

#if defined(__gfx1250__) && __has_builtin(__builtin_amdgcn_global_load_async_to_lds_b128)
#define USE_ASYNC 1
#else
#define USE_ASYNC 0
#endif

#if USE_ASYNC
typedef __attribute__((address_space(1))) v4i* gv4i_ptr;
typedef __attribute__((address_space(3))) v4i* lv4i_ptr;

__device__ __forceinline__ void async_cp16(const float* g, float* l) {
    // global_load_async_to_lds_b128: per-lane 16B HBM -> LDS, tracked by ASYNCcnt
    __builtin_amdgcn_global_load_async_to_lds_b128(
        (gv4i_ptr)(void*)g, (lv4i_ptr)(void*)l, 0, 0);
}

template <int N>
__device__ __forceinline__ void wait_async() {
#if __has_builtin(__builtin_amdgcn_s_wait_asynccnt)
    __builtin_amdgcn_s_wait_asynccnt(N);
#else
    asm volatile("s_wait_asynccnt %0" ::"i"(N) : "memory");
#endif
}
#endif

// s_total[r] = exp(log_s[r]*s_scale[r]) + exp(log_s_sigma[r])*noise[r]
__global__ __launch_bounds__(256) void svdlora_stotal(
    const float* __restrict__ log_s, const float* __restrict__ s_scale,
    const float* __restrict__ log_s_sigma, const float* __restrict__ noise,
    float* __restrict__ s_total, int R)
{
    int i = blockIdx.x * blockDim.x + threadIdx.x;
    if (i < R) {
        s_total[i] = expf(log_s[i] * s_scale[i]) + expf(log_s_sigma[i]) * noise[i];
    }
}

// Inner MMA over one staged BMxBK / BNxBK tile pair.
__device__ __forceinline__ void mma_tile(
    const float* __restrict__ Al, const float* __restrict__ Bl,
    v8f (&acc)[2][4], int waveM, int waveN, int row16, int khalf)
{
    #pragma unroll
    for (int k0 = 0; k0 < BK; k0 += 4) {
        v2f afrag[2], bfrag[4];
        #pragma unroll
        for (int i = 0; i < 2; ++i)
            afrag[i] = *(const v2f*)(&Al[(waveM * 32 + i * 16 + row16) * LDT + k0 + khalf]);
        #pragma unroll
        for (int j = 0; j < 4; ++j)
            bfrag[j] = *(const v2f*)(&Bl[(waveN * 64 + j * 16 + row16) * LDT + k0 + khalf]);
        #pragma unroll
        for (int i = 0; i < 2; ++i) {
            #pragma unroll
            for (int j = 0; j < 4; ++j) {
                acc[i][j] = __builtin_amdgcn_wmma_f32_16x16x4_f32(
                    false, afrag[i], false, bfrag[j],
                    (short)0, acc[i][j], false, false);
            }
        }
    }
}

// C[M,N] = (A[M,K] row-major) x (B[N,K] row-major)^T, optional per-column scale.
// 256 threads = 8 waves; block tile 128x128, K-step 32.
// Wave grid 4(M) x 2(N): each wave computes a 32x64 subtile = 2x4 WMMA frags.
__global__ __launch_bounds__(256) void svdlora_gemm_nt(
    const float* __restrict__ A,      // [M,K]
    const float* __restrict__ Bm,     // [N,K]
    float* __restrict__ C,            // [M,N]
    const float* __restrict__ scale,  // [N] or nullptr
    int M, int N, int K)
{
#if USE_ASYNC
    __shared__ float Alds[2][BM * LDT];
    __shared__ float Blds[2][BN * LDT];
#else
    __shared__ float Alds[1][BM * LDT];
    __shared__ float Blds[1][BN * LDT];
#endif

    const int tid   = threadIdx.x;
    const int lane  = tid & 31;
    const int wave  = tid >> 5;
    const int waveM = wave >> 1;          // 0..3 -> 32-row slice
    const int waveN = wave & 1;           // 0..1 -> 64-col slice
    const int row16 = lane & 15;
    const int khalf = (lane >> 4) << 1;   // 0 or 2 (ISA A/B frag K pairing)
    const int mhalf = (lane >> 4) << 3;   // 0 or 8 (ISA C/D frag M split)

    const int mBlock = blockIdx.y * BM;
    const int nBlock = blockIdx.x * BN;

    v8f acc[2][4] = {};
    const int numTiles = K / BK;

#if USE_ASYNC
    // ---- double-buffered async pipeline: tile kt+1 streams while tile kt computes ----
    #pragma unroll
    for (int it = 0; it < 4; ++it) {           // issue tile 0 -> buffer 0
        int idx = tid + it * 256;
        int r = idx >> 3, c = (idx & 7) << 2;
        async_cp16(A  + (size_t)(mBlock + r) * K + c, &Alds[0][r * LDT + c]);
        async_cp16(Bm + (size_t)(nBlock + r) * K + c, &Blds[0][r * LDT + c]);
    }
    for (int kt = 0; kt < numTiles; ++kt) {
        const int cur = kt & 1;
        if (kt + 1 < numTiles) {
            #pragma unroll
            for (int it = 0; it < 4; ++it) {   // prefetch next tile into other buffer
                int idx = tid + it * 256;
                int r = idx >> 3, c = (idx & 7) << 2;
                size_t koff = (size_t)(kt + 1) * BK + c;
                async_cp16(A  + (size_t)(mBlock + r) * K + koff, &Alds[cur ^ 1][r * LDT + c]);
                async_cp16(Bm + (size_t)(nBlock + r) * K + koff, &Blds[cur ^ 1][r * LDT + c]);
            }
            wait_async<8>();   // async completes in order: oldest 8 (= tile kt) are done
        } else {
            wait_async<0>();
        }
        __syncthreads();
        mma_tile(Alds[cur], Blds[cur], acc, waveM, waveN, row16, khalf);
        __syncthreads();
    }
#else
    // ---- fallback: synchronous staging, all 8 loads in flight before stores ----
    for (int kt = 0; kt < numTiles; ++kt) {
        v4f av[4], bv[4];
        #pragma unroll
        for (int it = 0; it < 4; ++it) {
            int idx = tid + it * 256;
            int r = idx >> 3, c = (idx & 7) << 2;
            av[it] = *(const v4f*)(A  + (size_t)(mBlock + r) * K + (size_t)kt * BK + c);
            bv[it] = *(const v4f*)(Bm + (size_t)(nBlock + r) * K + (size_t)kt * BK + c);
        }
        #pragma unroll
        for (int it = 0; it < 4; ++it) {
            int idx = tid + it * 256;
            int r = idx >> 3, c = (idx & 7) << 2;
            *(v4f*)(&Alds[0][r * LDT + c]) = av[it];
            *(v4f*)(&Blds[0][r * LDT + c]) = bv[it];
        }
        __syncthreads();
        mma_tile(Alds[0], Blds[0], acc, waveM, waveN, row16, khalf);
        __syncthreads();
    }
#endif

    // Epilogue: C/D frag layout -> lanes 0-15 hold N=row16, VGPR v holds M = v (+8 for hi half-wave)
    #pragma unroll
    for (int j = 0; j < 4; ++j) {
        const int n = nBlock + waveN * 64 + j * 16 + row16;
        const float sc = scale ? scale[n] : 1.0f;
        #pragma unroll
        for (int i = 0; i < 2; ++i) {
            const int mBase = mBlock + waveM * 32 + i * 16 + mhalf;
            #pragma unroll
            for (int v = 0; v < 8; ++v) {
                C[(size_t)(mBase + v) * N + n] = acc[i][j][v] * sc;
            }
        }
    }
}

extern "C" void kernel_launch(void* const* d_in, const int* in_sizes, int n_in,
                              void* d_out, int out_size, void* d_ws, size_t ws_size,
                              hipStream_t stream) {
    (void)n_in; (void)out_size; (void)ws_size;
    const float* x      = (const float*)d_in[0];
    const float* U      = (const float*)d_in[1];
    const float* Vh     = (const float*)d_in[2];
    const float* log_s  = (const float*)d_in[3];
    const float* s_scl  = (const float*)d_in[4];
    const float* log_sg = (const float*)d_in[5];
    const float* noise  = (const float*)d_in[6];
    float* out = (float*)d_out;

    const int RANK  = in_sizes[3];               // 4096
    const int D_IN  = in_sizes[2] / RANK;        // 4096
    const int D_OUT = in_sizes[1] / RANK;        // 4096
    const int Mrows = in_sizes[0] / D_IN;        // B*S = 8192

    float* s_total = (float*)d_ws;
    float* h = s_total + ((RANK + 1023) & ~1023);  // keep h 4KB-aligned

    svdlora_stotal<<<(RANK + 255) / 256, 256, 0, stream>>>(
        log_s, s_scl, log_sg, noise, s_total, RANK);

    dim3 blk(256);
    // GEMM1: h[m,r] = (x @ Vh^T)[m,r] * s_total[r]
    dim3 g1(RANK / BN, Mrows / BM);
    svdlora_gemm_nt<<<g1, blk, 0, stream>>>(x, Vh, h, s_total, Mrows, RANK, D_IN);
    // GEMM2: out[m,o] = h @ U^T
    dim3 g2(D_OUT / BN, Mrows / BM);
    svdlora_gemm_nt<<<g2, blk, 0, stream>>>(h, U, out, nullptr, Mrows, D_OUT, RANK);
}